// EfficientMLA_2138893714155
// MI455X (gfx1250) — compile-verified
//
#include <hip/hip_runtime.h>
#include <hip/hip_bf16.h>
#include <math.h>

// ---- problem constants ----
#define B_      8
#define S_PREV_ 4095
#define E_      2048
#define H_      16
#define HD_     128
#define LOW_    512
#define S_      4096          // S_PREV + 1
#define LN_ROPE 9.210340371976184f   // ln(10000)

typedef __attribute__((ext_vector_type(2))) float v2f;
typedef __attribute__((ext_vector_type(8))) float v8f;

// select cached row vs freshly-projected row (s == S_PREV_)
__device__ __forceinline__ const float* ckv_row_ptr(const float* cache, const float* fresh,
                                                    int b, int s) {
    return (s < S_PREV_) ? cache + ((size_t)b * S_PREV_ + s) * LOW_
                         : fresh + (size_t)b * LOW_;
}
__device__ __forceinline__ const float* kr_row_ptr(const float* cache, const float* fresh,
                                                   int b, int s) {
    return (s < S_PREV_) ? cache + ((size_t)b * S_PREV_ + s) * HD_
                         : fresh + (size_t)b * HD_;
}

// ---------------------------------------------------------------------------
// K1a: Cq = x@Wdq (B x E), ckv_new = x@Wdkv (B x LOW), kr_new = x@Wkr (B x HD)
// one thread per output element, K = E dot product
// ---------------------------------------------------------------------------
__global__ void k1a_proj(const float* __restrict__ x,
                         const float* __restrict__ Wdq,
                         const float* __restrict__ Wdkv,
                         const float* __restrict__ Wkr,
                         float* __restrict__ Cq,
                         float* __restrict__ ckv_new,
                         float* __restrict__ kr_new) {
    const int NPB = E_ + LOW_ + HD_;
    int idx = blockIdx.x * blockDim.x + threadIdx.x;
    if (idx >= B_ * NPB) return;
    int b = idx / NPB, r = idx % NPB;
    const float* xr = x + (size_t)b * E_;
    float acc = 0.f;
    if (r < E_) {
        for (int k = 0; k < E_; ++k) acc = fmaf(xr[k], Wdq[(size_t)k * E_ + r], acc);
        Cq[(size_t)b * E_ + r] = acc;
    } else if (r < E_ + LOW_) {
        int c = r - E_;
        for (int k = 0; k < E_; ++k) acc = fmaf(xr[k], Wdkv[(size_t)k * LOW_ + c], acc);
        ckv_new[(size_t)b * LOW_ + c] = acc;
    } else {
        int c = r - E_ - LOW_;
        for (int k = 0; k < E_; ++k) acc = fmaf(xr[k], Wkr[(size_t)k * HD_ + c], acc);
        kr_new[(size_t)b * HD_ + c] = acc;
    }
}

// ---------------------------------------------------------------------------
// K1b: Qc = Cq@Wuq (B x E);  Qr = rope(Cq@Wqr, pos = S-1) (B x HD)
// rope threads (64 per b) compute both halves of the rotated pair
// ---------------------------------------------------------------------------
__global__ void k1b_qproj(const float* __restrict__ Cq,
                          const float* __restrict__ Wuq,
                          const float* __restrict__ Wqr,
                          float* __restrict__ Qc,
                          float* __restrict__ Qr) {
    const int NPB = E_ + 64;
    int idx = blockIdx.x * blockDim.x + threadIdx.x;
    if (idx >= B_ * NPB) return;
    int b = idx / NPB, r = idx % NPB;
    const float* cq = Cq + (size_t)b * E_;
    if (r < E_) {
        float acc = 0.f;
        for (int k = 0; k < E_; ++k) acc = fmaf(cq[k], Wuq[(size_t)k * E_ + r], acc);
        Qc[(size_t)b * E_ + r] = acc;
    } else {
        int j = r - E_;  // 0..63
        float d1 = 0.f, d2 = 0.f;
        for (int k = 0; k < E_; ++k) {
            float c = cq[k];
            d1 = fmaf(c, Wqr[(size_t)k * HD_ + j], d1);
            d2 = fmaf(c, Wqr[(size_t)k * HD_ + j + 64], d2);
        }
        float inv = __expf(-(float)j * (LN_ROPE / 64.0f));
        float ang = (float)(S_ - 1) * inv;
        float sn, cs; __sincosf(ang, &sn, &cs);
        Qr[(size_t)b * HD_ + j]      = d1 * cs - d2 * sn;
        Qr[(size_t)b * HD_ + j + 64] = d1 * sn + d2 * cs;
    }
}

// ---------------------------------------------------------------------------
// K2: absorbed query  q_low[b,h,l] = sum_d Qc[b,h*HD+d] * Wuk[l, h*HD+d]
//     (Wuk = fused_upsample[:, :E], row stride 2E)
// ---------------------------------------------------------------------------
__global__ void k2_qlow(const float* __restrict__ Qc,
                        const float* __restrict__ fup,
                        float* __restrict__ q_low) {
    int idx = blockIdx.x * blockDim.x + threadIdx.x;  // B*H*LOW threads
    int b = idx / (H_ * LOW_);
    int rem = idx % (H_ * LOW_);
    int h = rem / LOW_, l = rem % LOW_;
    const float* q = Qc + (size_t)b * E_ + h * HD_;
    const float* w = fup + (size_t)l * (2 * E_) + h * HD_;
    float acc = 0.f;
#pragma unroll 4
    for (int d = 0; d < HD_; ++d) acc = fmaf(q[d], w[d], acc);
    q_low[((size_t)b * H_ + h) * LOW_ + l] = acc;
}

// ---------------------------------------------------------------------------
// K2b: rope-dot  sc_r[b,s] = Qr[b,:] . rope(kr[b,s,:], s)   (head-independent)
// ---------------------------------------------------------------------------
__global__ void k2b_ropescore(const float* __restrict__ kr_cache,
                              const float* __restrict__ kr_new,
                              const float* __restrict__ Qr,
                              float* __restrict__ sc_r) {
    int idx = blockIdx.x * blockDim.x + threadIdx.x;  // B*S threads
    int b = idx / S_, s = idx % S_;
    const float* kr = kr_row_ptr(kr_cache, kr_new, b, s);
    const float* q  = Qr + (size_t)b * HD_;
    float acc = 0.f;
    for (int j = 0; j < 64; ++j) {
        float inv = __expf(-(float)j * (LN_ROPE / 64.0f));
        float ang = (float)s * inv;
        float sn, cs; __sincosf(ang, &sn, &cs);
        float k1 = kr[j], k2 = kr[j + 64];
        acc = fmaf(q[j],      k1 * cs - k2 * sn, acc);
        acc = fmaf(q[j + 64], k1 * sn + k2 * cs, acc);
    }
    sc_r[(size_t)b * S_ + s] = acc;
}

// ---------------------------------------------------------------------------
// K3: scores[b,h,s] = (q_low[b,h,:] . ckv[b,s,:] + sc_r[b,s]) / 16
// WMMA f32 16x16x4: M=16 heads, N=16 seq positions, K=512 in steps of 4.
// One wave (32 lanes) per (b, seq-tile).
//   A lane layout: row m = lane&15, K pair = (lane<16 ? {0,1} : {2,3})
//   B lane layout: col n = lane&15 (seq),  K pair likewise -> contiguous b64
//   C/D: VGPR r -> rows r (lanes 0-15) / r+8 (lanes 16-31), col = lane&15
// ---------------------------------------------------------------------------
__global__ void __launch_bounds__(32)
k3_scores(const float* __restrict__ ckv_cache,
          const float* __restrict__ ckv_new,
          const float* __restrict__ q_low,
          const float* __restrict__ sc_r,
          float* __restrict__ scores) {
    int tile = blockIdx.x;                 // B * (S/16) tiles
    int b  = tile / (S_ / 16);
    int s0 = (tile % (S_ / 16)) * 16;
    int lane = threadIdx.x;
    int m  = lane & 15;
    int hi = lane >> 4;                    // 0: K{0,1}, 1: K{2,3}
    int s  = s0 + m;

    const float* arow = q_low + ((size_t)b * H_ + m) * LOW_ + hi * 2;
    const float* brow = ckv_row_ptr(ckv_cache, ckv_new, b, s) + hi * 2;

    __builtin_prefetch(brow, 0, 1);        // global_prefetch_b8 (gfx1250)
    __builtin_prefetch(arow, 0, 1);

    v8f c = {};
#pragma unroll 8
    for (int k = 0; k < LOW_; k += 4) {
        v2f a = *(const v2f*)(arow + k);
        v2f bb = *(const v2f*)(brow + k);
        c = __builtin_amdgcn_wmma_f32_16x16x4_f32(false, a, false, bb,
                                                  (short)0, c, false, false);
    }
    float add = sc_r[(size_t)b * S_ + s];
#pragma unroll
    for (int r = 0; r < 8; ++r) {
        int h = r + hi * 8;
        scores[((size_t)b * H_ + h) * S_ + s] = (c[r] + add) * (1.0f / 16.0f);
    }
}

// ---------------------------------------------------------------------------
// K4: softmax over S per (b,h), in place
// ---------------------------------------------------------------------------
__global__ void k4_softmax(float* __restrict__ scores) {
    __shared__ float red[256];
    float* row = scores + (size_t)blockIdx.x * S_;
    int tid = threadIdx.x;
    float m = -INFINITY;
    for (int i = tid; i < S_; i += 256) m = fmaxf(m, row[i]);
    red[tid] = m; __syncthreads();
    for (int off = 128; off > 0; off >>= 1) {
        if (tid < off) red[tid] = fmaxf(red[tid], red[tid + off]);
        __syncthreads();
    }
    float bm = red[0]; __syncthreads();
    float sum = 0.f;
    for (int i = tid; i < S_; i += 256) {
        float e = __expf(row[i] - bm);
        row[i] = e; sum += e;
    }
    red[tid] = sum; __syncthreads();
    for (int off = 128; off > 0; off >>= 1) {
        if (tid < off) red[tid] += red[tid + off];
        __syncthreads();
    }
    float inv = 1.0f / red[0];
    for (int i = tid; i < S_; i += 256) row[i] *= inv;
}

// ---------------------------------------------------------------------------
// K5: latent output  o_low[b,h,l] = sum_s attn[b,h,s] * ckv[b,s,l]
// WMMA f32 16x16x4: M=16 heads, N=16 latent cols, K=4096 seq in steps of 4.
// ckv (67MB) is L2-resident after K3 on the 192MB L2.
// ---------------------------------------------------------------------------
__global__ void __launch_bounds__(32)
k5_olow(const float* __restrict__ ckv_cache,
        const float* __restrict__ ckv_new,
        const float* __restrict__ attn,
        float* __restrict__ o_low) {
    int tile = blockIdx.x;                 // B * (LOW/16) tiles
    int b  = tile / (LOW_ / 16);
    int l0 = (tile % (LOW_ / 16)) * 16;
    int lane = threadIdx.x;
    int m  = lane & 15;
    int hi = lane >> 4;
    int l  = l0 + m;

    const float* arow = attn + ((size_t)b * H_ + m) * S_ + hi * 2;

    v8f c = {};
#pragma unroll 4
    for (int k = 0; k < S_; k += 4) {
        v2f a = *(const v2f*)(arow + k);
        int s0r = k + hi * 2;
        v2f bb;
        bb.x = ckv_row_ptr(ckv_cache, ckv_new, b, s0r)[l];
        bb.y = ckv_row_ptr(ckv_cache, ckv_new, b, s0r + 1)[l];
        c = __builtin_amdgcn_wmma_f32_16x16x4_f32(false, a, false, bb,
                                                  (short)0, c, false, false);
    }
#pragma unroll
    for (int r = 0; r < 8; ++r) {
        int h = r + hi * 8;
        o_low[((size_t)b * H_ + h) * LOW_ + l] = c[r];
    }
}

// ---------------------------------------------------------------------------
// K6: up-project  o_flat[b,e] = sum_l o_low[b, e/HD, l] * Wuv[l, e]
//     (Wuv = fused_upsample[:, E:], row stride 2E)
// ---------------------------------------------------------------------------
__global__ void k6_upv(const float* __restrict__ o_low,
                       const float* __restrict__ fup,
                       float* __restrict__ o_flat) {
    int idx = blockIdx.x * blockDim.x + threadIdx.x;  // B*E threads
    int b = idx / E_, e = idx % E_;
    int h = e / HD_;
    const float* ol = o_low + ((size_t)b * H_ + h) * LOW_;
    const float* w  = fup + E_ + e;                   // column e of Wuv
    float acc = 0.f;
#pragma unroll 4
    for (int l = 0; l < LOW_; ++l) acc = fmaf(ol[l], w[(size_t)l * (2 * E_)], acc);
    o_flat[(size_t)b * E_ + e] = acc;
}

// ---------------------------------------------------------------------------
// K7: out = o_flat @ Wo   (B x E)
// ---------------------------------------------------------------------------
__global__ void k7_out(const float* __restrict__ o_flat,
                       const float* __restrict__ Wo,
                       float* __restrict__ out) {
    int idx = blockIdx.x * blockDim.x + threadIdx.x;  // B*E threads
    int b = idx / E_, n = idx % E_;
    const float* o = o_flat + (size_t)b * E_;
    float acc = 0.f;
#pragma unroll 4
    for (int e = 0; e < E_; ++e) acc = fmaf(o[e], Wo[(size_t)e * E_ + n], acc);
    out[(size_t)b * E_ + n] = acc;
}

// ---------------------------------------------------------------------------
extern "C" void kernel_launch(void* const* d_in, const int* in_sizes, int n_in,
                              void* d_out, int out_size, void* d_ws, size_t ws_size,
                              hipStream_t stream) {
    const float* x    = (const float*)d_in[0];
    const float* ckvc = (const float*)d_in[1];
    const float* krc  = (const float*)d_in[2];
    const float* Wdq  = (const float*)d_in[3];
    const float* Wuq  = (const float*)d_in[4];
    const float* Wqr  = (const float*)d_in[5];
    const float* Wdkv = (const float*)d_in[6];
    const float* Wkr  = (const float*)d_in[7];
    const float* fup  = (const float*)d_in[8];
    const float* Wo   = (const float*)d_in[9];
    float* out = (float*)d_out;

    // workspace carve-up (floats)
    float* ws      = (float*)d_ws;
    float* Cq      = ws;                        // B*E
    float* Qc      = Cq      + (size_t)B_ * E_; // B*E
    float* Qr      = Qc      + (size_t)B_ * E_; // B*HD
    float* ckv_new = Qr      + (size_t)B_ * HD_;// B*LOW
    float* kr_new  = ckv_new + (size_t)B_ * LOW_;// B*HD
    float* q_low   = kr_new  + (size_t)B_ * HD_; // B*H*LOW
    float* sc_r    = q_low   + (size_t)B_ * H_ * LOW_;  // B*S
    float* scores  = sc_r    + (size_t)B_ * S_;         // B*H*S
    float* o_low   = scores  + (size_t)B_ * H_ * S_;    // B*H*LOW
    float* o_flat  = o_low   + (size_t)B_ * H_ * LOW_;  // B*E

    {   // K1a: x-projections
        int total = B_ * (E_ + LOW_ + HD_);
        k1a_proj<<<(total + 255) / 256, 256, 0, stream>>>(x, Wdq, Wdkv, Wkr,
                                                          Cq, ckv_new, kr_new);
    }
    {   // K1b: Qc + roped Qr
        int total = B_ * (E_ + 64);
        k1b_qproj<<<(total + 255) / 256, 256, 0, stream>>>(Cq, Wuq, Wqr, Qc, Qr);
    }
    // K2: absorbed query (Wuk folded into q)
    k2_qlow<<<(B_ * H_ * LOW_) / 256, 256, 0, stream>>>(Qc, fup, q_low);
    // K2b: head-independent rope scores
    k2b_ropescore<<<(B_ * S_) / 256, 256, 0, stream>>>(krc, kr_new, Qr, sc_r);
    // K3: latent-space attention scores (WMMA)
    k3_scores<<<B_ * (S_ / 16), 32, 0, stream>>>(ckvc, ckv_new, q_low, sc_r, scores);
    // K4: softmax
    k4_softmax<<<B_ * H_, 256, 0, stream>>>(scores);
    // K5: latent-space output accumulation (WMMA)
    k5_olow<<<B_ * (LOW_ / 16), 32, 0, stream>>>(ckvc, ckv_new, scores, o_low);
    // K6: Wuv up-projection
    k6_upv<<<(B_ * E_) / 256, 256, 0, stream>>>(o_low, fup, o_flat);
    // K7: output projection
    k7_out<<<(B_ * E_) / 256, 256, 0, stream>>>(o_flat, Wo, out);
}